// CTCLoss_4346506903683
// MI455X (gfx1250) — compile-verified
//
#include <hip/hip_runtime.h>
#include <cstdint>
#include <math.h>

// Problem constants from the reference: B=32, T=1024, V=512, L=128.
#define TPAR    1024
#define BATCH   32
#define VOCAB   512
#define LMAX    128
#define SEXT    (2*LMAX + 1)      // 257 extended-lattice states
#define CHUNK   8                 // frames per async-staged chunk (16 KB)
#define NCHUNKS (TPAR / CHUNK)
#define NTHREADS 288              // 9 wave32s; threads [0,257) own a lattice state
#define NEGINF  (-1e30f)

// Low 32 bits of a generic pointer to __shared__ == LDS byte offset (flat LDS aperture).
__device__ __forceinline__ uint32_t lds_off_u32(const void* p) {
    return (uint32_t)(uintptr_t)p;
}

__global__ __launch_bounds__(NTHREADS)
void ctc_alpha_kernel(const float* __restrict__ log_probs,      // (B,T,V) log-softmaxed
                      const int*   __restrict__ targets,        // (B,L)
                      const int*   __restrict__ input_lengths,  // (B,)
                      const int*   __restrict__ target_lengths, // (B,)
                      float*       __restrict__ per_batch)      // (B,) scratch
{
    __shared__ float rowbuf[2][CHUNK * VOCAB];   // double-buffered frame rows (2 x 16 KB)
    __shared__ float alpha[2][SEXT + 3];         // ping-pong alpha lattice

    const int b    = blockIdx.x;
    const int tid  = threadIdx.x;
    const int s    = tid;
    const int ilen = input_lengths[b];
    const int tlen = target_lengths[b];

    // Extended label symbol for state s, and whether the s-2 skip transition is legal.
    int  ext  = 0;       // blank for even s
    bool skip = false;
    if (s < SEXT && (s & 1)) {
        const int k = (s - 1) >> 1;
        ext = targets[b * LMAX + k];
        const int prev = (k >= 1) ? targets[b * LMAX + k - 1] : -1;
        skip = (s >= 2) && (ext != 0) && (ext != prev);   // s>=2 && nonblank && differs
    }

    const char* gbase = (const char*)log_probs + (size_t)b * TPAR * VOCAB * sizeof(float);

    // Wave 0 streams one CHUNK of frame rows (16 KB) into rowbuf[bufsel] via the
    // CDNA5 async data-mover path: 32 x global_load_async_to_lds_b128 (512 B each).
    auto issue_chunk = [&](int c, int bufsel) {
        if (tid < 32) {
            const uint64_t g0     = (uint64_t)gbase + (uint64_t)c * (CHUNK * VOCAB * 4);
            const uint32_t l0     = lds_off_u32(&rowbuf[bufsel][0]);
            const uint32_t lane16 = (uint32_t)tid * 16u;
            #pragma unroll
            for (int k = 0; k < (CHUNK * VOCAB * 4) / 512; ++k) {
                const uint64_t ga = g0 + (uint32_t)(k * 512) + lane16;
                const uint32_t la = l0 + (uint32_t)(k * 512) + lane16;
                asm volatile("global_load_async_to_lds_b128 %0, %1, off"
                             :: "v"(la), "v"(ga) : "memory");
            }
        }
    };

    // Prefetch chunk 0 and wait for it.
    issue_chunk(0, 0);
    asm volatile("s_wait_asynccnt 0x0" ::: "memory");
    __syncthreads();

    // t = 0 initialization: alpha[0]=lp(blank), alpha[1]=lp(first label), rest -inf.
    int cur = 0;
    if (s < SEXT) {
        float a0;
        if      (s == 0) a0 = rowbuf[0][0];
        else if (s == 1) a0 = rowbuf[0][ext];
        else             a0 = NEGINF;
        alpha[0][s] = a0;
    }
    __syncthreads();

    for (int t = 1; t < TPAR; ++t) {
        const int c  = t / CHUNK;
        const int tt = t % CHUNK;
        const int rb = c & 1;

        // On entering a chunk (and at t==1 for chunk 0), kick off the next chunk's DMA.
        if ((tt == 0) || (t == 1)) {
            const int cn = c + 1;
            if (cn < NCHUNKS) issue_chunk(cn, cn & 1);
        }

        if (s < SEXT) {
            const float a  = alpha[cur][s];
            const float a1 = (s >= 1) ? alpha[cur][s - 1] : NEGINF;
            const float a2 = skip     ? alpha[cur][s - 2] : NEGINF;
            const float m  = fmaxf(fmaxf(a, a1), a2);
            const float lp = rowbuf[rb][tt * VOCAB + ext];
            const float nv = m + logf(expf(a - m) + expf(a1 - m) + expf(a2 - m)) + lp;
            alpha[cur ^ 1][s] = (t < ilen) ? nv : a;   // freeze past input_length
        }

        // Before the first use of the next chunk's buffer, its DMA must be complete.
        if (tt == CHUNK - 1) {
            asm volatile("s_wait_asynccnt 0x0" ::: "memory");
        }
        __syncthreads();
        cur ^= 1;
    }

    if (tid == 0) {
        const int   e  = 2 * tlen;                  // final blank state
        const float x  = alpha[cur][e];
        const float y  = alpha[cur][e - 1];
        const float m  = fmaxf(x, y);
        const float ll = m + logf(expf(x - m) + expf(y - m));
        const float nll = -ll;
        per_batch[b] = (nll < 1e29f) ? (nll / (float)tlen) : 0.0f;  // zero_infinity
    }
}

// Deterministic mean over B (avoids float-atomic ordering nondeterminism).
__global__ void ctc_reduce_kernel(const float* __restrict__ per_batch,
                                  float* __restrict__ out)
{
    if (threadIdx.x == 0) {
        float acc = 0.0f;
        for (int i = 0; i < BATCH; ++i) acc += per_batch[i];
        out[0] = acc / (float)BATCH;
    }
}

extern "C" void kernel_launch(void* const* d_in, const int* in_sizes, int n_in,
                              void* d_out, int out_size, void* d_ws, size_t ws_size,
                              hipStream_t stream) {
    const float* log_probs      = (const float*)d_in[0];
    const int*   targets        = (const int*)  d_in[1];
    const int*   input_lengths  = (const int*)  d_in[2];
    const int*   target_lengths = (const int*)  d_in[3];
    float* per_batch = (float*)d_ws;   // 32 floats of scratch

    ctc_alpha_kernel<<<BATCH, NTHREADS, 0, stream>>>(
        log_probs, targets, input_lengths, target_lengths, per_batch);
    ctc_reduce_kernel<<<1, 32, 0, stream>>>(per_batch, (float*)d_out);
}